// PolicyModule_44813688767173
// MI455X (gfx1250) — compile-verified
//
#include <hip/hip_runtime.h>
#include <math.h>

typedef __attribute__((ext_vector_type(2))) float v2f;
typedef __attribute__((ext_vector_type(8))) float v8f;

#define NEG_SLOPE 0.2f

// ---------------------------------------------------------------------------
// Generic fill
// ---------------------------------------------------------------------------
__global__ void fill_kernel(float* __restrict__ p, float v, int n) {
    int t = blockIdx.x * blockDim.x + threadIdx.x;
    if (t < n) p[t] = v;
}

// ---------------------------------------------------------------------------
// WMMA f32 GEMM: D[M,Ncols] = A[M,K] * B[K,Ncols], all row-major fp32.
// One wave (32 lanes) computes one 16x16 tile via V_WMMA_F32_16X16X4_F32.
// A 16x4 layout: lanes 0-15 rows M=0..15 hold K={k,k+1}; lanes 16-31 hold K={k+2,k+3}.
// B 4x16 layout: symmetric (lane&15 = column, lane>>4 selects K pair).
// C/D 16x16 f32: VGPR i -> (M = i + 8*(lane>>4), N = lane&15).
// ---------------------------------------------------------------------------
__global__ __launch_bounds__(32) void wmma_gemm_f32(const float* __restrict__ A,
                                                    const float* __restrict__ B,
                                                    float* __restrict__ D,
                                                    int K, int ldb) {
    const int lane = threadIdx.x;        // 0..31
    const int half = lane >> 4;          // 0 or 1
    const int l15  = lane & 15;
    const int r0   = blockIdx.x * 16;
    const int c0   = blockIdx.y * 16;
    const int arow = r0 + l15;
    const int bcol = c0 + l15;

    v8f acc = {};
    for (int k = 0; k < K; k += 4) {
        const int ka = k + 2 * half;
        v2f a, b;
        a.x = A[(size_t)arow * K + ka];
        a.y = A[(size_t)arow * K + ka + 1];
        b.x = B[(size_t)ka * ldb + bcol];
        b.y = B[(size_t)(ka + 1) * ldb + bcol];
        acc = __builtin_amdgcn_wmma_f32_16x16x4_f32(
            /*neg_a=*/false, a, /*neg_b=*/false, b,
            /*c_mod=*/(short)0, acc, /*reuse_a=*/false, /*reuse_b=*/false);
    }
#pragma unroll
    for (int i = 0; i < 8; ++i) {
        D[(size_t)(r0 + i + 8 * half) * ldb + c0 + l15] = acc[i];
    }
}

// ---------------------------------------------------------------------------
// a_src[n,h] = dot(h[n,h,:], att_src[h,:]) ; same for a_dst
// ---------------------------------------------------------------------------
__global__ void attn_dots(const float* __restrict__ h,
                          const float* __restrict__ att_s,
                          const float* __restrict__ att_d,
                          float* __restrict__ asrc, float* __restrict__ adst,
                          int N, int Hd, int C) {
    int t = blockIdx.x * blockDim.x + threadIdx.x;
    if (t >= N * Hd) return;
    int n = t / Hd, hh = t - n * Hd;
    const float* hp = h + ((size_t)n * Hd + hh) * C;
    const float* s  = att_s + (size_t)hh * C;
    const float* d  = att_d + (size_t)hh * C;
    float as = 0.f, ad = 0.f;
    for (int c = 0; c < C; ++c) { float v = hp[c]; as += v * s[c]; ad += v * d[c]; }
    asrc[t] = as; adst[t] = ad;
}

__device__ __forceinline__ void edge_nodes(const int* __restrict__ ei,
                                           int e, int E, int* src, int* dst) {
    if (e < E) { *src = ei[e]; *dst = ei[E + e]; }
    else       { *src = e - E; *dst = e - E; }    // self-loop
}

__device__ __forceinline__ float lrelu(float v) {
    return v > 0.f ? v : NEG_SLOPE * v;
}

__device__ __forceinline__ void atomicMaxF(float* addr, float val) {
    int* ai = (int*)addr;
    int old = __float_as_int(*addr);
    while (__int_as_float(old) < val) {
        int assumed = old;
        old = atomicCAS(ai, assumed, __float_as_int(val));
        if (old == assumed) break;
    }
}

// segment max of leaky-relu edge logits into m[dst,h]
__global__ void edge_max(const int* __restrict__ ei, int E, int EE, int Hd,
                         const float* __restrict__ asrc,
                         const float* __restrict__ adst,
                         float* __restrict__ m) {
    int t = blockIdx.x * blockDim.x + threadIdx.x;
    if (t >= EE * Hd) return;
    int e = t / Hd, hh = t - e * Hd;
    int s, d; edge_nodes(ei, e, E, &s, &d);
    float v = lrelu(asrc[s * Hd + hh] + adst[d * Hd + hh]);
    atomicMaxF(&m[d * Hd + hh], v);
}

// ee = exp(e - m[dst]); denom[dst] += ee; stash ee per (edge,head)
__global__ void edge_exp(const int* __restrict__ ei, int E, int EE, int Hd,
                         const float* __restrict__ asrc,
                         const float* __restrict__ adst,
                         const float* __restrict__ m,
                         float* __restrict__ den, float* __restrict__ ee) {
    int t = blockIdx.x * blockDim.x + threadIdx.x;
    if (t >= EE * Hd) return;
    int e = t / Hd, hh = t - e * Hd;
    int s, d; edge_nodes(ei, e, E, &s, &d);
    float v  = lrelu(asrc[s * Hd + hh] + adst[d * Hd + hh]);
    float ev = __expf(v - m[d * Hd + hh]);
    ee[t] = ev;
    atomicAdd(&den[d * Hd + hh], ev);
}

// agg[dst,h,:] += h[src,h,:] * alpha
__global__ void edge_agg(const int* __restrict__ ei, int E, int EE, int Hd, int C,
                         const float* __restrict__ ee,
                         const float* __restrict__ den,
                         const float* __restrict__ hfeat,
                         float* __restrict__ agg) {
    int t = blockIdx.x * blockDim.x + threadIdx.x;
    if (t >= EE * Hd) return;
    int e = t / Hd, hh = t - e * Hd;
    int s, d; edge_nodes(ei, e, E, &s, &d);
    float alpha = ee[t] / (den[d * Hd + hh] + 1e-16f);
    const float* hp = hfeat + ((size_t)s * Hd + hh) * C;
    float* ap = agg + ((size_t)d * Hd + hh) * C;
    for (int c = 0; c < C; ++c) atomicAdd(&ap[c], hp[c] * alpha);
}

// out[n,c] = relu(mean_h(agg[n,h,c]) + bias[c])
__global__ void head_mean_bias_relu(const float* __restrict__ agg,
                                    const float* __restrict__ bias,
                                    float* __restrict__ out,
                                    int N, int Hd, int C) {
    int t = blockIdx.x * blockDim.x + threadIdx.x;
    if (t >= N * C) return;
    int n = t / C, c = t - n * C;
    float a = 0.f;
    for (int hh = 0; hh < Hd; ++hh) a += agg[((size_t)n * Hd + hh) * C + c];
    a = a / (float)Hd + bias[c];
    out[t] = a > 0.f ? a : 0.f;
}

// global mean pool accumulation
__global__ void pool_kernel(const float* __restrict__ h2,
                            const int* __restrict__ batch,
                            float* __restrict__ gsum, float* __restrict__ gcnt,
                            int N, int C) {
    int t = blockIdx.x * blockDim.x + threadIdx.x;
    if (t >= N * C) return;
    int n = t / C, c = t - n * C;
    int g = batch[n];
    atomicAdd(&gsum[g * C + c], h2[t]);
    if (c == 0) atomicAdd(&gcnt[g], 1.0f);
}

// two 32->16->1 sigmoid heads; out[0:G]=halt, out[G:2G]=cont
__global__ void heads_kernel(const float* __restrict__ gsum,
                             const float* __restrict__ gcnt,
                             const float* __restrict__ cW1, const float* __restrict__ cb1,
                             const float* __restrict__ cW2, const float* __restrict__ cb2,
                             const float* __restrict__ hW1, const float* __restrict__ hb1,
                             const float* __restrict__ hW2, const float* __restrict__ hb2,
                             float* __restrict__ out, int G) {
    int g = blockIdx.x * blockDim.x + threadIdx.x;
    if (g >= G) return;
    float cnt = gcnt[g]; if (cnt < 1.f) cnt = 1.f;
    float emb[32];
    for (int c = 0; c < 32; ++c) emb[c] = gsum[g * 32 + c] / cnt;
    float hid[16];
    // continue head
    for (int j = 0; j < 16; ++j) {
        float a = cb1[j];
        for (int c = 0; c < 32; ++c) a += emb[c] * cW1[c * 16 + j];
        hid[j] = a > 0.f ? a : 0.f;
    }
    float oc = cb2[0];
    for (int j = 0; j < 16; ++j) oc += hid[j] * cW2[j];
    float cont = 1.f / (1.f + __expf(-oc));
    // halt head
    for (int j = 0; j < 16; ++j) {
        float a = hb1[j];
        for (int c = 0; c < 32; ++c) a += emb[c] * hW1[c * 16 + j];
        hid[j] = a > 0.f ? a : 0.f;
    }
    float oh = hb2[0];
    for (int j = 0; j < 16; ++j) oh += hid[j] * hW2[j];
    float halt = 1.f / (1.f + __expf(-oh));
    out[g] = halt;          // tuple element 0
    out[G + g] = cont;      // tuple element 1
}

// ---------------------------------------------------------------------------
// Launch
// ---------------------------------------------------------------------------
static inline int cdiv(long long a, int b) { return (int)((a + b - 1) / b); }

extern "C" void kernel_launch(void* const* d_in, const int* in_sizes, int n_in,
                              void* d_out, int out_size, void* d_ws, size_t ws_size,
                              hipStream_t stream) {
    const float* x     = (const float*)d_in[0];
    const int*   ei    = (const int*)  d_in[1];
    const int*   batch = (const int*)  d_in[2];
    const float* W1    = (const float*)d_in[3];
    const float* as1   = (const float*)d_in[4];
    const float* ad1   = (const float*)d_in[5];
    const float* b1    = (const float*)d_in[6];
    const float* W2    = (const float*)d_in[7];
    const float* as2   = (const float*)d_in[8];
    const float* ad2   = (const float*)d_in[9];
    const float* b2    = (const float*)d_in[10];
    const float* cW1   = (const float*)d_in[11];
    const float* cb1   = (const float*)d_in[12];
    const float* cW2   = (const float*)d_in[13];
    const float* cb2   = (const float*)d_in[14];
    const float* hW1   = (const float*)d_in[15];
    const float* hb1   = (const float*)d_in[16];
    const float* hW2   = (const float*)d_in[17];
    const float* hb2   = (const float*)d_in[18];

    const int N  = in_sizes[0] / 128;   // 20000
    const int E  = in_sizes[1] / 2;     // 320000
    const int EE = E + N;               // with self-loops
    const int G  = 64;

    float* ws = (float*)d_ws;
    size_t off = 0;
    auto alloc = [&](size_t n) { float* p = ws + off; off += n; return p; };

    float* hpre1 = alloc((size_t)N * 256);
    float* asrc1 = alloc((size_t)N * 4);
    float* adst1 = alloc((size_t)N * 4);
    float* m1    = alloc((size_t)N * 4);
    float* den1  = alloc((size_t)N * 4);
    float* ee1   = alloc((size_t)EE * 4);
    float* agg1  = alloc((size_t)N * 4 * 64);
    float* h1    = alloc((size_t)N * 64);
    float* hpre2 = alloc((size_t)N * 64);
    float* asrc2 = alloc((size_t)N * 2);
    float* adst2 = alloc((size_t)N * 2);
    float* m2    = alloc((size_t)N * 2);
    float* den2  = alloc((size_t)N * 2);
    float* ee2   = alloc((size_t)EE * 2);
    float* agg2  = alloc((size_t)N * 2 * 32);
    float* h2    = alloc((size_t)N * 32);
    float* gsum  = alloc((size_t)G * 32);
    float* gcnt  = alloc((size_t)G);

    const int TB = 256;

    // ---------------- Layer 1: heads=4, C=64, K=128, cols=256 ----------------
    wmma_gemm_f32<<<dim3(N / 16, 256 / 16), 32, 0, stream>>>(x, W1, hpre1, 128, 256);
    attn_dots<<<cdiv((long long)N * 4, TB), TB, 0, stream>>>(hpre1, as1, ad1, asrc1, adst1, N, 4, 64);
    fill_kernel<<<cdiv((long long)N * 4, TB), TB, 0, stream>>>(m1, -3.0e38f, N * 4);
    fill_kernel<<<cdiv((long long)N * 4, TB), TB, 0, stream>>>(den1, 0.f, N * 4);
    fill_kernel<<<cdiv((long long)N * 4 * 64, TB), TB, 0, stream>>>(agg1, 0.f, N * 4 * 64);
    edge_max<<<cdiv((long long)EE * 4, TB), TB, 0, stream>>>(ei, E, EE, 4, asrc1, adst1, m1);
    edge_exp<<<cdiv((long long)EE * 4, TB), TB, 0, stream>>>(ei, E, EE, 4, asrc1, adst1, m1, den1, ee1);
    edge_agg<<<cdiv((long long)EE * 4, TB), TB, 0, stream>>>(ei, E, EE, 4, 64, ee1, den1, hpre1, agg1);
    head_mean_bias_relu<<<cdiv((long long)N * 64, TB), TB, 0, stream>>>(agg1, b1, h1, N, 4, 64);

    // ---------------- Layer 2: heads=2, C=32, K=64, cols=64 ------------------
    wmma_gemm_f32<<<dim3(N / 16, 64 / 16), 32, 0, stream>>>(h1, W2, hpre2, 64, 64);
    attn_dots<<<cdiv((long long)N * 2, TB), TB, 0, stream>>>(hpre2, as2, ad2, asrc2, adst2, N, 2, 32);
    fill_kernel<<<cdiv((long long)N * 2, TB), TB, 0, stream>>>(m2, -3.0e38f, N * 2);
    fill_kernel<<<cdiv((long long)N * 2, TB), TB, 0, stream>>>(den2, 0.f, N * 2);
    fill_kernel<<<cdiv((long long)N * 2 * 32, TB), TB, 0, stream>>>(agg2, 0.f, N * 2 * 32);
    edge_max<<<cdiv((long long)EE * 2, TB), TB, 0, stream>>>(ei, E, EE, 2, asrc2, adst2, m2);
    edge_exp<<<cdiv((long long)EE * 2, TB), TB, 0, stream>>>(ei, E, EE, 2, asrc2, adst2, m2, den2, ee2);
    edge_agg<<<cdiv((long long)EE * 2, TB), TB, 0, stream>>>(ei, E, EE, 2, 32, ee2, den2, hpre2, agg2);
    head_mean_bias_relu<<<cdiv((long long)N * 32, TB), TB, 0, stream>>>(agg2, b2, h2, N, 2, 32);

    // ---------------- Pool + heads ----------------
    fill_kernel<<<1, 256, 0, stream>>>(gsum, 0.f, G * 32);
    fill_kernel<<<1, 64, 0, stream>>>(gcnt, 0.f, G);
    pool_kernel<<<cdiv((long long)N * 32, TB), TB, 0, stream>>>(h2, batch, gsum, gcnt, N, 32);
    heads_kernel<<<2, 32, 0, stream>>>(gsum, gcnt, cW1, cb1, cW2, cb2,
                                       hW1, hb1, hW2, hb2, (float*)d_out, G);
}